// GaussianHFCFilter_59450937311645
// MI455X (gfx1250) — compile-verified
//
#include <hip/hip_runtime.h>
#include <hip/hip_bf16.h>

typedef float v2f __attribute__((ext_vector_type(2)));
typedef float v8f __attribute__((ext_vector_type(8)));

#define TILE      64
#define HALO      11     // 23-tap filter radius
#define IN_ROWS   96     // 86 live rows padded to 6x16 for WMMA row tiles
#define IN_PITCH  88     // 86 live cols + pad (zeroed)
#define H_ROWS    96
#define H_PITCH   72

// ---------------------------------------------------------------------------
// Exact selection of BOTH adjacent ranks (rankBase, rankBase+1) per plane in a
// single workgroup: 3-level radix select (11/11/10 bits) on the monotone
// float->uint map. While both ranks share a bit-prefix (typical), one LDS
// histogram serves both; after divergence a second histogram takes over.
// quantize!=0 applies trunc(v*256)/256 on the fly (the reference's "temp").
// Streaming scans use float4 (global_load_b128).
// ---------------------------------------------------------------------------
__global__ __launch_bounds__(1024)
void rank_pair_select_kernel(const float* __restrict__ src, int planeN, int quantize,
                             int rankBase, float* __restrict__ out)
{
    const int plane = blockIdx.x;
    const float4* p4 = (const float4*)(src + (size_t)plane * (size_t)planeN);
    const int n4 = planeN >> 2;

    __shared__ unsigned hist0[2048];
    __shared__ unsigned hist1[2048];
    __shared__ unsigned sB[4];   // prefix0, rank0, prefix1, rank1

    unsigned prefix0 = 0u, prefix1 = 0u;
    unsigned rank0 = (unsigned)rankBase, rank1 = (unsigned)rankBase + 1u;
    unsigned hiMask = 0u;
    bool diverged = false;

    for (int pass = 0; pass < 3; ++pass) {
        const int shift = (pass == 0) ? 21 : (pass == 1) ? 10 : 0;
        const int bins  = (pass == 2) ? 1024 : 2048;

        for (int i = threadIdx.x; i < 2048; i += blockDim.x) { hist0[i] = 0u; hist1[i] = 0u; }
        __syncthreads();

        for (int i = threadIdx.x; i < n4; i += blockDim.x) {
            float4 v4 = p4[i];
            float vv[4] = { v4.x, v4.y, v4.z, v4.w };
#pragma unroll
            for (int j = 0; j < 4; ++j) {
                float v = vv[j];
                if (quantize) v = truncf(v * 256.0f) * (1.0f / 256.0f);
                unsigned u = __float_as_uint(v);
                unsigned m = (u & 0x80000000u) ? ~u : (u | 0x80000000u);
                unsigned bin = (m >> shift) & (unsigned)(bins - 1);
                if ((m & hiMask) == prefix0) atomicAdd(&hist0[bin], 1u);
                if (diverged && (m & hiMask) == prefix1) atomicAdd(&hist1[bin], 1u);
            }
        }
        __syncthreads();

        if (threadIdx.x == 0) {
            unsigned cum = 0; int b = 0;
            for (; b < bins; ++b) { unsigned c = hist0[b]; if (cum + c > rank0) break; cum += c; }
            if (b >= bins) b = bins - 1;
            sB[0] = prefix0 | ((unsigned)b << shift);
            sB[1] = rank0 - cum;

            const unsigned* h = diverged ? hist1 : hist0;
            cum = 0; b = 0;
            for (; b < bins; ++b) { unsigned c = h[b]; if (cum + c > rank1) break; cum += c; }
            if (b >= bins) b = bins - 1;
            sB[2] = prefix1 | ((unsigned)b << shift);
            sB[3] = rank1 - cum;
        }
        __syncthreads();
        prefix0 = sB[0]; rank0 = sB[1];
        prefix1 = sB[2]; rank1 = sB[3];
        hiMask |= (unsigned)(bins - 1) << shift;
        diverged = (prefix0 != prefix1);
        __syncthreads();
    }

    if (threadIdx.x == 0) {
        unsigned m = prefix0;
        unsigned u = (m & 0x80000000u) ? (m & 0x7fffffffu) : ~m;
        out[plane * 2 + 0] = __uint_as_float(u);
        m = prefix1;
        u = (m & 0x80000000u) ? (m & 0x7fffffffu) : ~m;
        out[plane * 2 + 1] = __uint_as_float(u);
    }
}

// ---------------------------------------------------------------------------
// Fused: median fill -> separable 23-tap Gaussian (replicate pad) -> res.
// One 64x64 output tile per workgroup (256 threads = 8 wave32).
// Both separable passes run as V_WMMA_F32_16X16X4_F32 chains:
//   h-pass: D(16x16) = A(image patch 16xK) x B(tap band Kx16), K=40
//   v-pass: D(16x16) = A(tap band 16xK)   x B(column patch Kx16)
// The band operand is a pure function of the lane -> lives in 10 v2f regs.
// ---------------------------------------------------------------------------
__global__ __launch_bounds__(256)
void blur_res_kernel(const float* __restrict__ x, const float* __restrict__ mask,
                     const float* __restrict__ medpair, float* __restrict__ res,
                     int C, int H, int W)
{
    __shared__ float sIn[IN_ROWS * IN_PITCH];
    __shared__ float sH [H_ROWS * H_PITCH];
    __shared__ float sGW[23];

    const int plane = blockIdx.z;
    const int n     = plane / C;
    const int r0    = blockIdx.y * TILE;
    const int c0    = blockIdx.x * TILE;
    const int tid   = threadIdx.x;

    // Gaussian taps (sigma = 9), normalized
    if (tid < 23) {
        float d = (float)tid - 11.0f;
        sGW[tid] = expf(-(d * d) / (2.0f * 9.0f * 9.0f));
    }
    __syncthreads();
    if (tid == 0) {
        float s = 0.0f;
        for (int i = 0; i < 23; ++i) s += sGW[i];
        float inv = 1.0f / s;
        for (int i = 0; i < 23; ++i) sGW[i] *= inv;
    }
    // zero entire staging tile (padding rows/cols must be finite zeros)
    for (int i = tid; i < IN_ROWS * IN_PITCH; i += 256) sIn[i] = 0.0f;
    __syncthreads();

    const float med = 0.5f * (medpair[plane * 2] + medpair[plane * 2 + 1]) + 0.2f;
    const size_t planeOff = (size_t)plane * H * W;
    const size_t maskOff  = (size_t)n * H * W;

    // load 86x86 halo region, clamped (replicate pad), median-filled
    for (int i = tid; i < 86 * 86; i += 256) {
        int rr = i / 86, cc = i % 86;
        int gr = min(max(r0 - HALO + rr, 0), H - 1);
        int gc = min(max(c0 - HALO + cc, 0), W - 1);
        float xv = x[planeOff + (size_t)gr * W + gc];
        float mv = mask[maskOff + (size_t)gr * W + gc];
        sIn[rr * IN_PITCH + cc] = mv * xv + (1.0f - mv) * med;
    }
    __syncthreads();

    const int wave  = tid >> 5;
    const int lane  = tid & 31;
    const int nIdx  = lane & 15;   // M for A-operand, N for B/D-operands
    const int khalf = lane >> 4;   // K half-select per ISA f32 layouts

    // Tap band chunks: band[kk] supplies both the h-pass B operand
    // (B[j][c] = w[j-c]) and the v-pass A operand (A[r][j] = w[j-r]).
    v2f band[10];
#pragma unroll
    for (int kk = 0; kk < 10; ++kk) {
        int j0 = kk * 4 + khalf * 2;
        int d0 = j0 - nIdx, d1 = j0 + 1 - nIdx;
        band[kk].x = (d0 >= 0 && d0 < 23) ? sGW[d0] : 0.0f;
        band[kk].y = (d1 >= 0 && d1 < 23) ? sGW[d1] : 0.0f;
    }

    // -------- horizontal pass: 24 tiles (6 row-tiles x 4 col-tiles) --------
    for (int t = wave; t < 24; t += 8) {
        int rowBase = (t >> 2) * 16;          // sIn/sH row tile
        int colBase = (t & 3) * 16;           // output col tile
        v8f acc = {};
#pragma unroll
        for (int kk = 0; kk < 10; ++kk) {
            int j0 = kk * 4 + khalf * 2;
            const float* src = &sIn[(rowBase + nIdx) * IN_PITCH + colBase + j0];
            v2f a; a.x = src[0]; a.y = src[1];
            acc = __builtin_amdgcn_wmma_f32_16x16x4_f32(
                      false, a, false, band[kk], (short)0, acc, false, false);
        }
#pragma unroll
        for (int i = 0; i < 8; ++i)
            sH[(rowBase + khalf * 8 + i) * H_PITCH + colBase + nIdx] = acc[i];
    }
    __syncthreads();

    // -------- vertical pass + res: 16 tiles (4x4) --------
    for (int t = wave; t < 16; t += 8) {
        int rowBase = (t >> 2) * 16;          // output row tile
        int colBase = (t & 3) * 16;
        v8f acc = {};
#pragma unroll
        for (int kk = 0; kk < 10; ++kk) {
            int j0 = kk * 4 + khalf * 2;
            v2f b;
            b.x = sH[(rowBase + j0    ) * H_PITCH + colBase + nIdx];
            b.y = sH[(rowBase + j0 + 1) * H_PITCH + colBase + nIdx];
            acc = __builtin_amdgcn_wmma_f32_16x16x4_f32(
                      false, band[kk], false, b, (short)0, acc, false, false);
        }
#pragma unroll
        for (int i = 0; i < 8; ++i) {
            int m  = khalf * 8 + i;
            int gr = r0 + rowBase + m;
            int gc = c0 + colBase + nIdx;
            float filled = sIn[(HALO + rowBase + m) * IN_PITCH + HALO + colBase + nIdx];
            res[planeOff + (size_t)gr * W + gc] = 4.0f * (filled - acc[i]);
        }
    }
}

// ---------------------------------------------------------------------------
// out = (res - lo) / (hi - lo) * mask, with linearly interpolated percentiles.
// float4 streams; planeN is a multiple of 4 so a vector never crosses planes.
// ---------------------------------------------------------------------------
__global__ __launch_bounds__(256)
void finalize_kernel(const float4* __restrict__ res4, const float4* __restrict__ mask4,
                     const float* __restrict__ lopair, const float* __restrict__ hipair,
                     float4* __restrict__ out4, int C, int planeN, size_t total4,
                     float fracLo, float fracHi)
{
    size_t idx = (size_t)blockIdx.x * blockDim.x + threadIdx.x;
    if (idx >= total4) return;
    const int planeN4 = planeN >> 2;
    int plane  = (int)(idx / (size_t)planeN4);
    int within = (int)(idx % (size_t)planeN4);
    int nn     = plane / C;

    float l0 = lopair[plane * 2], l1 = lopair[plane * 2 + 1];
    float h0 = hipair[plane * 2], h1 = hipair[plane * 2 + 1];
    float lo = l0 + fracLo * (l1 - l0);
    float hi = h0 + fracHi * (h1 - h0);
    float inv = 1.0f / (hi - lo);

    float4 r  = res4[idx];
    float4 mv = mask4[(size_t)nn * planeN4 + within];
    float4 o;
    o.x = (r.x - lo) * inv * mv.x;
    o.y = (r.y - lo) * inv * mv.y;
    o.z = (r.z - lo) * inv * mv.z;
    o.w = (r.w - lo) * inv * mv.w;
    out4[idx] = o;
}

extern "C" void kernel_launch(void* const* d_in, const int* in_sizes, int n_in,
                              void* d_out, int out_size, void* d_ws, size_t ws_size,
                              hipStream_t stream)
{
    const float* x    = (const float*)d_in[0];
    const float* mask = (const float*)d_in[1];
    const int H = 512, W = 512;
    const int planeN = H * W;
    const int planes = in_sizes[0] / planeN;   // B*C = 96
    const int Bb     = in_sizes[1] / planeN;   // 32
    const int C      = planes / Bb;            // 3

    float* res     = (float*)d_ws;
    float* medpair = res + (size_t)in_sizes[0];
    float* lopair  = medpair + planes * 2;
    float* hipair  = lopair  + planes * 2;

    // 1) exact median pair on raw x (one WG per plane resolves both ranks)
    int rankMed = (planeN - 1) / 2;                       // 131071 (pair with +1)
    rank_pair_select_kernel<<<planes, 1024, 0, stream>>>(x, planeN, 0, rankMed, medpair);

    // 2) fused fill + separable Gaussian (WMMA) + res
    dim3 bGrid(W / TILE, H / TILE, planes);
    blur_res_kernel<<<bGrid, 256, 0, stream>>>(x, mask, medpair, res, C, H, W);

    // 3) percentile rank pairs on trunc-quantized res
    double qLo = 0.03 * (double)(planeN - 1);
    double qHi = 0.97 * (double)(planeN - 1);
    int rLo = (int)qLo, rHi = (int)qHi;
    rank_pair_select_kernel<<<planes, 1024, 0, stream>>>(res, planeN, 1, rLo, lopair);
    rank_pair_select_kernel<<<planes, 1024, 0, stream>>>(res, planeN, 1, rHi, hipair);

    // 4) normalize + re-mask (float4 streams)
    size_t total4 = (size_t)out_size >> 2;
    int blocks = (int)((total4 + 255) / 256);
    finalize_kernel<<<blocks, 256, 0, stream>>>((const float4*)res, (const float4*)mask,
                                                lopair, hipair, (float4*)d_out,
                                                C, planeN, total4,
                                                (float)(qLo - rLo), (float)(qHi - rHi));
}